// TDecM_42984032699116
// MI455X (gfx1250) — compile-verified
//
#include <hip/hip_runtime.h>

typedef __attribute__((ext_vector_type(2))) float v2f;
typedef __attribute__((ext_vector_type(8))) float v8f;

#define NFRM 5
#define NB   8
#define NC   64
#define HWT  16384        // 128*128
#define NK   320          // NFRM*NC
#define CH   64           // pixel chunks per batch
#define PIXC 256          // HWT/CH pixels per chunk

// ---------------------------------------------------------------------------
// Pass A: per (b, chunk) block. GEMM origin = W * x via v_wmma_f32_16x16x4_f32,
// then accumulate per-(c,i) dot(origin, d_i) and sum(d_i^2) partials.
// Block = 256 threads = 8 waves. Wave w: channel tile (w&3)*16, pixel half w>>2.
// Each wave: 8 tiles of 16 pixels (128 pixels), 16 channels.
// ---------------------------------------------------------------------------
__global__ __launch_bounds__(256) void passA_kernel(
    const float* __restrict__ inp,      // (5, 8, 64, 16384)
    const float* __restrict__ Wt,       // (64, 320)
    const float* __restrict__ biasv,    // (64,)
    float* __restrict__ partials)       // (B*CH, 512)
{
    __shared__ float lacc[NC * 4 * 2];  // [c][i][dot,sq] = 512 floats
    const int tid = threadIdx.x;
    for (int j = tid; j < 512; j += 256) lacc[j] = 0.0f;
    __syncthreads();

    const int blk   = blockIdx.x;
    const int b     = blk / CH;
    const int chunk = blk % CH;
    const int wave  = tid >> 5;
    const int lane  = tid & 31;
    const int l15   = lane & 15;
    const int hi    = lane >> 4;          // 0 or 1 (half-wave)
    const int c_tile = (wave & 3) * 16;   // 16-channel tile
    const int pix0   = chunk * PIXC + (wave >> 2) * 128;

    // Per-lane bias for the 8 accumulator rows (channel = c_tile + hi*8 + r)
    float biasr[8];
#pragma unroll
    for (int r = 0; r < 8; ++r) biasr[r] = biasv[c_tile + hi * 8 + r];

    const v8f vzero = {0.f, 0.f, 0.f, 0.f, 0.f, 0.f, 0.f, 0.f};
    v8f acc[8];
#pragma unroll
    for (int t = 0; t < 8; ++t) acc[t] = vzero;

    // ---- GEMM: origin[c_tile+0..15][pix0..pix0+127] over K=320 ----
#pragma unroll 1
    for (int f = 0; f < NFRM; ++f) {
        const float* xf = inp + ((f * NB + b) * NC) * HWT;
        // A fragments (weights) for this frame: 16 K-steps of 4.
        // Lane layout: a.x -> K = base + hi*2, a.y -> K = base + hi*2 + 1
        v2f afr[16];
#pragma unroll
        for (int s = 0; s < 16; ++s) {
            const int kk = f * 64 + s * 4 + hi * 2;
            afr[s] = *(const v2f*)(Wt + (c_tile + l15) * NK + kk);
        }
#pragma unroll
        for (int t = 0; t < 8; ++t) {
            const int p = pix0 + t * 16 + l15;
            const float* xp = xf + p;
#pragma unroll
            for (int s = 0; s < 16; ++s) {
                const int cA = s * 4 + hi * 2;   // channel within frame
                v2f bfr;
                bfr.x = xp[cA * HWT];
                bfr.y = xp[(cA + 1) * HWT];
                acc[t] = __builtin_amdgcn_wmma_f32_16x16x4_f32(
                    false, afr[s], false, bfr, (short)0, acc[t], false, false);
            }
        }
    }

    // ---- dot(origin, d_i) and sum(d_i^2) reductions ----
    const float* lastb = inp + ((4 * NB + b) * NC) * HWT;
#pragma unroll 1
    for (int t = 0; t < 8; ++t) {
        const int p = pix0 + t * 16 + l15;
        float lv[8], ov[8];
#pragma unroll
        for (int r = 0; r < 8; ++r) {
            const int ch = c_tile + hi * 8 + r;
            lv[r] = lastb[ch * HWT + p];
            ov[r] = acc[t][r] + biasr[r];
        }
#pragma unroll
        for (int i = 0; i < 4; ++i) {
            const float* fb = inp + ((i * NB + b) * NC) * HWT + p;
#pragma unroll
            for (int r = 0; r < 8; ++r) {
                const int ch = c_tile + hi * 8 + r;
                const float d = lv[r] - fb[ch * HWT];
                float pd = ov[r] * d;
                float ps = d * d;
#pragma unroll
                for (int off = 1; off < 16; off <<= 1) {
                    pd += __shfl_xor(pd, off, 16);
                    ps += __shfl_xor(ps, off, 16);
                }
                if (l15 == 0) {
                    const int idx = (ch * 4 + i) * 2;
                    atomicAdd(&lacc[idx],     pd);
                    atomicAdd(&lacc[idx + 1], ps);
                }
            }
        }
    }
    __syncthreads();
    float* pb = partials + (size_t)blk * 512;
    for (int j = tid; j < 512; j += 256) pb[j] = lacc[j];
}

// ---------------------------------------------------------------------------
// Pass B: reduce chunk partials -> coefficients -> fused 320-tap vector v[b,:]
// v[b, f*64+c] = -ow0[c]*coef_f (f<4);  v[b, 256+c] = ow0[c]*sum(coef) + ow1[c]
// ---------------------------------------------------------------------------
__global__ void coef_kernel(const float* __restrict__ partials,
                            const float* __restrict__ out_w,   // (1,128)
                            float* __restrict__ v)             // (8,320)
{
    const int b = blockIdx.x;
    const int c = threadIdx.x;   // 64 threads
    const float ow0 = out_w[c];
    const float ow1 = out_w[64 + c];
    float csum = 0.0f;
#pragma unroll
    for (int i = 0; i < 4; ++i) {
        float dot = 0.0f, sq = 0.0f;
        for (int ch = 0; ch < CH; ++ch) {
            const size_t base = (size_t)(b * CH + ch) * 512 + (c * 4 + i) * 2;
            dot += partials[base];
            sq  += partials[base + 1];
        }
        float n = sqrtf(sq);
        n = fmaxf(n, 1e-12f);
        const float coef = dot / (n * n);
        csum += coef;
        v[b * NK + i * 64 + c] = -ow0 * coef;
    }
    v[b * NK + 256 + c] = ow0 * csum + ow1;
}

// ---------------------------------------------------------------------------
// Pass C: y[b,p] = out_b + sum_k v[b,k] * x[b,p,k]  (streaming, memory-bound)
// ---------------------------------------------------------------------------
__global__ __launch_bounds__(256) void passC_kernel(
    const float* __restrict__ inp,
    const float* __restrict__ v,
    const float* __restrict__ out_b,
    float* __restrict__ y)
{
    __shared__ float vl[NK];
    const int tid = threadIdx.x;
    const int b = blockIdx.x >> 6;                 // 64 blocks per batch
    const int p = (blockIdx.x & 63) * 256 + tid;
    for (int j = tid; j < NK; j += 256) vl[j] = v[b * NK + j];
    __syncthreads();

    float acc = out_b[0];
#pragma unroll 1
    for (int f = 0; f < NFRM; ++f) {
        const float* xp = inp + ((f * NB + b) * NC) * HWT + p;
        const float* vf = vl + f * 64;
#pragma unroll 8
        for (int c = 0; c < NC; ++c)
            acc = fmaf(vf[c], xp[c * HWT], acc);
    }
    y[(size_t)b * HWT + p] = acc;
}

// ---------------------------------------------------------------------------
extern "C" void kernel_launch(void* const* d_in, const int* in_sizes, int n_in,
                              void* d_out, int out_size, void* d_ws, size_t ws_size,
                              hipStream_t stream) {
    const float* inp      = (const float*)d_in[0];  // (5,8,64,128,128)
    const float* origin_w = (const float*)d_in[1];  // (64,320)
    const float* origin_b = (const float*)d_in[2];  // (64,)
    const float* out_w    = (const float*)d_in[3];  // (1,128)
    const float* out_b    = (const float*)d_in[4];  // (1,)
    float* y = (float*)d_out;                       // (8,1,128,128)

    float* partials = (float*)d_ws;                 // 512 blocks * 512 floats = 1 MB
    float* v        = partials + 512 * 512;         // 8*320 floats

    passA_kernel<<<NB * CH, 256, 0, stream>>>(inp, origin_w, origin_b, partials);
    coef_kernel <<<NB, NC, 0, stream>>>(partials, out_w, v);
    passC_kernel<<<NB * 64, 256, 0, stream>>>(inp, v, out_b, y);
}